// PNN_62156766707845
// MI455X (gfx1250) — compile-verified
//
#include <hip/hip_runtime.h>
#include <stdint.h>

typedef __attribute__((ext_vector_type(2))) float v2f;
typedef __attribute__((ext_vector_type(8))) float v8f;

#define B_TOT   16384
#define F_NUM   39
#define V_NUM   100000
#define E_NUM   16
#define K_TOT   624        // F * E
#define CHUNK_F 13
#define CHUNK_K 208        // CHUNK_F * E
#define N_CHUNK 3
#define KS      212        // LDS row stride (floats): 848 B = 16B-aligned, conflict-free
#define XVS     41         // Xv tile stride (odd -> conflict-free scalar reads)
#define XSTR    34         // MLP transpose buffer stride
#define WAVES   4
#define AGATHER ((16 * CHUNK_F * 4) / 32)   // 26 async B128 per chunk per lane

// generic pointer -> LDS byte offset (address_space(3) pointers are LDS offsets)
__device__ __forceinline__ uint32_t lds_off(const void* p) {
    return (uint32_t)(uintptr_t)(__attribute__((address_space(3))) const void*)p;
}

__device__ __forceinline__ v8f wmma_f32(v2f a, v2f b, v8f c) {
    // 8 args: (neg_a, A, neg_b, B, c_mod, C, reuse_a, reuse_b)
    return __builtin_amdgcn_wmma_f32_16x16x4_f32(
        false, a, false, b, (short)0, c, false, false);
}

// Issue 26 async global->LDS B128 copies: raw 16-float embedding rows for
// 16 batch rows x 13 fields (chunk c) into this wave's LDS buffer.
__device__ __forceinline__ void issue_gather(
    int c, float* buf, const int* __restrict__ Xi,
    const float* __restrict__ emb, int rowbase, int lane)
{
    #pragma unroll
    for (int t = 0; t < AGATHER; ++t) {
        int g    = t * 32 + lane;
        int q    = g & 3;            // which float4 of the 16-float row
        int pair = g >> 2;
        int rl   = pair & 15;        // local batch row
        int fl   = pair >> 4;        // local field
        int f    = c * CHUNK_F + fl;
        int idx  = Xi[(rowbase + rl) * F_NUM + f];
        const float* src = emb + ((size_t)f * V_NUM + idx) * E_NUM + q * 4;
        uint32_t dst = lds_off(buf + rl * KS + fl * 16 + q * 4);
        asm volatile("global_load_async_to_lds_b128 %0, %1, off"
                     :: "v"(dst), "v"(src) : "memory");
    }
}

__global__ __launch_bounds__(WAVES * 32) void pnn_fused(
    const int*   __restrict__ Xi,   const float* __restrict__ Xv,
    const float* __restrict__ emb,  const float* __restrict__ w_first,
    const float* __restrict__ w_inner,
    const float* __restrict__ l1W,  const float* __restrict__ l1b,
    const float* __restrict__ l2W,  const float* __restrict__ l2b,
    const float* __restrict__ lastW,const float* __restrict__ lastb,
    float* __restrict__ out)
{
    __shared__ float lds[WAVES][2 * 16 * KS + 16 * XVS];

    const int tid  = threadIdx.x;
    const int lane = tid & 31;
    const int wave = tid >> 5;
    const int m    = lane & 15;          // row within tile (A layout), column (B/C layouts)
    const int khi  = (lane >> 4) << 1;   // +2 K offset for lanes 16..31
    const int nlo  = m;
    const int nhi  = m + 16;
    const int rowbase = blockIdx.x * (WAVES * 16) + wave * 16;

    float* buf0 = &lds[wave][0];
    float* buf1 = &lds[wave][16 * KS];
    float* xvt  = &lds[wave][2 * 16 * KS];

    // ---- stage the Xv tile (16 rows x 39 fields) into LDS ----
    for (int t = lane; t < 16 * F_NUM; t += 32) {
        int rl = t / F_NUM;
        int f  = t - rl * F_NUM;
        xvt[rl * XVS + f] = Xv[(rowbase + rl) * F_NUM + f];
    }

    // ---- pipelined: async-gather chunk c+1 while WMMAing chunk c ----
    issue_gather(0, buf0, Xi, emb, rowbase, lane);

    v8f cf_lo = {}, cf_hi = {}, cs_lo = {}, cs_hi = {};

    for (int c = 0; c < N_CHUNK; ++c) {
        float* cur = (c & 1) ? buf1 : buf0;
        float* nxt = (c & 1) ? buf0 : buf1;
        if (c + 1 < N_CHUNK) {
            issue_gather(c + 1, nxt, Xi, emb, rowbase, lane);
            // in-order completion: <=26 outstanding => chunk c fully in LDS
            asm volatile("s_wait_asynccnt 26" ::: "memory");
        } else {
            asm volatile("s_wait_asynccnt 0" ::: "memory");
        }

        for (int fl = 0; fl < CHUNK_F; ++fl) {
            float xv = xvt[m * XVS + c * CHUNK_F + fl];
            #pragma unroll
            for (int e4 = 0; e4 < E_NUM; e4 += 4) {
                int ks = fl * 16 + e4;
                int kg = c * CHUNK_K + ks + khi;
                v2f a = *(const v2f*)&cur[m * KS + ks + khi];
                a *= xv;                           // fold Xv scale into A
                v2f bf_lo = *(const v2f*)(w_first + nlo * K_TOT + kg);
                v2f bf_hi = *(const v2f*)(w_first + nhi * K_TOT + kg);
                v2f bi_lo = *(const v2f*)(w_inner + nlo * K_TOT + kg);
                v2f bi_hi = *(const v2f*)(w_inner + nhi * K_TOT + kg);
                cf_lo = wmma_f32(a, bf_lo, cf_lo);
                cf_hi = wmma_f32(a, bf_hi, cf_hi);
                cs_lo = wmma_f32(a, bi_lo, cs_lo);
                cs_hi = wmma_f32(a, bi_hi, cs_hi);
            }
        }
    }

    // ---- x = first + s*s (elementwise in C/D layout) ----
    v8f x_lo = cf_lo + cs_lo * cs_lo;
    v8f x_hi = cf_hi + cs_hi * cs_hi;

    const int rA = (lane < 16) ? 0 : 8;
    float* ebuf = buf0;

    // ---- transpose x (16x32) into LDS as next A operand ----
    #pragma unroll
    for (int v = 0; v < 8; ++v) {
        ebuf[(rA + v) * XSTR + nlo] = x_lo[v];
        ebuf[(rA + v) * XSTR + nhi] = x_hi[v];
    }

    // ---- layer 1: relu(x @ l1W^T + l1b), K = 32 ----
    v8f y_lo = {}, y_hi = {};
    #pragma unroll
    for (int ks = 0; ks < 32; ks += 4) {
        v2f a   = *(const v2f*)&ebuf[m * XSTR + ks + khi];
        v2f blo = *(const v2f*)(l1W + nlo * 32 + ks + khi);
        v2f bhi = *(const v2f*)(l1W + nhi * 32 + ks + khi);
        y_lo = wmma_f32(a, blo, y_lo);
        y_hi = wmma_f32(a, bhi, y_hi);
    }
    {
        float b0 = l1b[nlo], b1 = l1b[nhi];
        #pragma unroll
        for (int v = 0; v < 8; ++v) {
            ebuf[(rA + v) * XSTR + nlo] = fmaxf(y_lo[v] + b0, 0.0f);
            ebuf[(rA + v) * XSTR + nhi] = fmaxf(y_hi[v] + b1, 0.0f);
        }
    }

    // ---- layer 2: relu(y @ l2W^T + l2b), K = 32 ----
    v8f z_lo = {}, z_hi = {};
    #pragma unroll
    for (int ks = 0; ks < 32; ks += 4) {
        v2f a   = *(const v2f*)&ebuf[m * XSTR + ks + khi];
        v2f blo = *(const v2f*)(l2W + nlo * 32 + ks + khi);
        v2f bhi = *(const v2f*)(l2W + nhi * 32 + ks + khi);
        z_lo = wmma_f32(a, blo, z_lo);
        z_hi = wmma_f32(a, bhi, z_hi);
    }

    // ---- last layer + row sum ----
    float wA = lastW[nlo], wB = lastW[nhi];
    float b0 = l2b[nlo],   b1 = l2b[nhi];
    float lb = lastb[0];
    float p[8];
    #pragma unroll
    for (int v = 0; v < 8; ++v) {
        p[v] = fmaxf(z_lo[v] + b0, 0.0f) * wA
             + fmaxf(z_hi[v] + b1, 0.0f) * wB;
    }
    #pragma unroll
    for (int off = 8; off >= 1; off >>= 1) {
        #pragma unroll
        for (int v = 0; v < 8; ++v) p[v] += __shfl_xor(p[v], off, 16);
    }
    if ((lane & 15) == 0) {
        int r0 = rowbase + ((lane >> 4) ? 8 : 0);
        #pragma unroll
        for (int v = 0; v < 8; ++v) out[r0 + v] = p[v] + lb;
    }
}

extern "C" void kernel_launch(void* const* d_in, const int* in_sizes, int n_in,
                              void* d_out, int out_size, void* d_ws, size_t ws_size,
                              hipStream_t stream) {
    const int*   Xi  = (const int*)  d_in[0];
    const float* Xv  = (const float*)d_in[1];
    const float* emb = (const float*)d_in[2];
    const float* wf  = (const float*)d_in[3];
    const float* wi  = (const float*)d_in[4];
    const float* l1W = (const float*)d_in[5];
    const float* l1b = (const float*)d_in[6];
    const float* l2W = (const float*)d_in[7];
    const float* l2b = (const float*)d_in[8];
    const float* lW  = (const float*)d_in[9];
    const float* lbp = (const float*)d_in[10];
    float* out = (float*)d_out;

    dim3 grid(B_TOT / (WAVES * 16));
    dim3 block(WAVES * 32);
    hipLaunchKernelGGL(pnn_fused, grid, block, 0, stream,
                       Xi, Xv, emb, wf, wi, l1W, l1b, l2W, l2b, lW, lbp, out);
}